// HebbNet_33337536151735
// MI455X (gfx1250) — compile-verified
//
#include <hip/hip_runtime.h>

#define T_ 32
#define B_ 32
#define NX 512
#define NH 512
#define NY 64

typedef __attribute__((ext_vector_type(2))) float v2f;
typedef __attribute__((ext_vector_type(8))) float v8f;

__device__ __forceinline__ float sigmoidf_(float a) {
    return 1.0f / (1.0f + __expf(-a));
}

// -----------------------------------------------------------------------------
// Kernel 1: XW1[m,h] = sum_k X[m,k]*w1[h,k],  m = t*B+b  (X flat [1024,512])
// One wave per 16x16 tile; K stepped by 4 via V_WMMA_F32_16X16X4_F32 (full f32).
// A frag (16x4): lanes 0-15 -> K={k,k+1}, lanes 16-31 -> K={k+2,k+3}; row = lane&15.
// B frag (4x16): same K split; col = lane&15; B[k,n] = w1[n,k] (contiguous pair).
// -----------------------------------------------------------------------------
__global__ void k_xw1(const float* __restrict__ X, const float* __restrict__ W1,
                      float* __restrict__ Out) {
    const unsigned wid  = threadIdx.x >> 5;
    const unsigned lane = threadIdx.x & 31;
    const unsigned tile  = blockIdx.x * 8 + wid;   // 2048 tiles total
    const unsigned tileM = tile >> 5;              // 64 M-tiles
    const unsigned tileN = tile & 31;              // 32 N-tiles

    const unsigned kh  = lane >> 4;                // lane half selects K offset
    const unsigned l15 = lane & 15;
    const unsigned mrow = tileM * 16 + l15;
    const unsigned ncol = tileN * 16 + l15;

    v8f acc = {};
    for (unsigned k = 0; k < NX; k += 4) {
        const unsigned kk = k + kh * 2;
        v2f a = *(const v2f*)(X  + mrow * NX + kk);   // A[m,kk], A[m,kk+1]
        v2f b = *(const v2f*)(W1 + ncol * NX + kk);   // B[kk,n]=w1[n,kk], B[kk+1,n]
        acc = __builtin_amdgcn_wmma_f32_16x16x4_f32(false, a, false, b,
                                                    (short)0, acc, false, false);
    }
    // C/D: VGPR r, lanes 0-15 -> M=r; lanes 16-31 -> M=r+8; N = lane&15
    const unsigned mbase = tileM * 16 + kh * 8;
#pragma unroll
    for (int r = 0; r < 8; ++r)
        Out[(mbase + r) * NH + ncol] = acc[r];
}

// -----------------------------------------------------------------------------
// Kernel 2: per-batch input Gram matrix  G[b,t,s] = x[t,b,:] . x[s,b,:]
// 128 tiles: (b, ti, si) with ti,si in {0,1}; one wave per tile.
// -----------------------------------------------------------------------------
__global__ void k_gram(const float* __restrict__ X, float* __restrict__ G) {
    const unsigned lane = threadIdx.x & 31;
    const unsigned tile = blockIdx.x;
    const unsigned b  = tile >> 2;
    const unsigned ti = (tile >> 1) & 1;
    const unsigned si = tile & 1;

    const unsigned kh  = lane >> 4;
    const unsigned l15 = lane & 15;
    const unsigned trow = ti * 16 + l15;
    const unsigned scol = si * 16 + l15;

    const float* Xa = X + (trow * B_ + b) * NX;    // row t of X_b
    const float* Xb = X + (scol * B_ + b) * NX;    // row s of X_b

    v8f acc = {};
    for (unsigned k = 0; k < NX; k += 4) {
        const unsigned kk = k + kh * 2;
        v2f a  = *(const v2f*)(Xa + kk);
        v2f bb = *(const v2f*)(Xb + kk);
        acc = __builtin_amdgcn_wmma_f32_16x16x4_f32(false, a, false, bb,
                                                    (short)0, acc, false, false);
    }
    const unsigned tbase = ti * 16 + kh * 8;
#pragma unroll
    for (int r = 0; r < 8; ++r)
        G[b * (T_ * T_) + (tbase + r) * T_ + scol] = acc[r];
}

// -----------------------------------------------------------------------------
// Kernel 3: sequential Hebbian scan (critical path).
// h_t[b,h] = sigmoid( XW1[t,b,h] + b1[h] + sum_{s<t} c[t,s]*h_s[b,h] )
// with c[t,s] = eta * lam^(t-1-s) * G[b,t,s].  Each (b,h) lane is independent:
// history kept in 32 registers (fully unrolled), coefficients in 4KB LDS.
// H buffer holds XW1 on entry and is overwritten in place with the hiddens.
// -----------------------------------------------------------------------------
__global__ void k_scan(const float* __restrict__ G, const float* __restrict__ B1,
                       const float* __restrict__ lam_p, const float* __restrict__ eta_p,
                       float* __restrict__ H) {
    __shared__ float coef[T_ * T_];
    const unsigned b = blockIdx.x;
    const unsigned h = threadIdx.x;                 // 512 threads = 16 waves

    const float lamc = fminf(lam_p[0], 1.0f);       // torch clamps lam to max=1.0
    const float eta  = eta_p[0];

    for (unsigned e = h; e < T_ * T_; e += NH) {
        const unsigned t = e >> 5, s = e & 31;
        float c = 0.0f;
        if (s < t) {
            c = eta * G[b * (T_ * T_) + e];
            for (unsigned i = s + 1; i < t; ++i) c *= lamc;   // lam^(t-1-s)
        }
        coef[e] = c;
    }
    __syncthreads();

    const float b1v = B1[h];
    float hist[T_];
#pragma unroll
    for (int t = 0; t < T_; ++t) {
        float a = H[(t * B_ + b) * NH + h] + b1v;
#pragma unroll
        for (int s = 0; s < t; ++s)
            a = fmaf(coef[t * T_ + s], hist[s], a);
        const float hv = sigmoidf_(a);
        hist[t] = hv;
        H[(t * B_ + b) * NH + h] = hv;              // hiddens output [T,B,Nh,1]
    }
}

// -----------------------------------------------------------------------------
// Kernel 4: Y = sigmoid(H @ w2 + b2)   [1024,512]x[512,64] via f32 WMMA.
// w2 is stored [Nh,Ny] so B[k,n] = w2[k*NY+n] (two b32 loads per step).
// -----------------------------------------------------------------------------
__global__ void k_out(const float* __restrict__ H, const float* __restrict__ W2,
                      const float* __restrict__ B2, float* __restrict__ Y) {
    const unsigned wid  = threadIdx.x >> 5;
    const unsigned lane = threadIdx.x & 31;
    const unsigned tile  = blockIdx.x * 8 + wid;   // 256 tiles
    const unsigned tileM = tile >> 2;              // 64 M-tiles
    const unsigned tileN = tile & 3;               // 4 N-tiles

    const unsigned kh  = lane >> 4;
    const unsigned l15 = lane & 15;
    const unsigned mrow = tileM * 16 + l15;
    const unsigned ncol = tileN * 16 + l15;

    v8f acc = {};
    for (unsigned k = 0; k < NH; k += 4) {
        const unsigned kk = k + kh * 2;
        v2f a = *(const v2f*)(H + mrow * NH + kk);
        v2f b;
        b.x = W2[kk * NY + ncol];
        b.y = W2[(kk + 1) * NY + ncol];
        acc = __builtin_amdgcn_wmma_f32_16x16x4_f32(false, a, false, b,
                                                    (short)0, acc, false, false);
    }
    const float bias = B2[ncol];
    const unsigned mbase = tileM * 16 + kh * 8;
#pragma unroll
    for (int r = 0; r < 8; ++r)
        Y[(mbase + r) * NY + ncol] = sigmoidf_(acc[r] + bias);
}

extern "C" void kernel_launch(void* const* d_in, const int* in_sizes, int n_in,
                              void* d_out, int out_size, void* d_ws, size_t ws_size,
                              hipStream_t stream) {
    const float* x   = (const float*)d_in[0];   // [T,B,Nx]
    const float* w1  = (const float*)d_in[1];   // [Nh,Nx]
    const float* b1  = (const float*)d_in[2];   // [Nh,1]
    const float* w2  = (const float*)d_in[3];   // [Nh,Ny]
    const float* b2  = (const float*)d_in[4];   // [Ny]
    const float* lam = (const float*)d_in[5];   // scalar
    const float* eta = (const float*)d_in[6];   // scalar

    float* hid = (float*)d_out;                 // [T*B,Nh]: XW1, then hiddens in place
    float* y   = (float*)d_out + T_ * B_ * NH;  // [T*B,Ny]
    float* G   = (float*)d_ws;                  // [B,T,T] = 128 KB scratch

    k_xw1 <<<256, 256, 0, stream>>>(x, w1, hid);
    k_gram<<<128,  32, 0, stream>>>(x, G);
    k_scan<<<B_,  NH, 0, stream>>>(G, b1, lam, eta, hid);
    k_out <<<32,  256, 0, stream>>>(hid, w2, b2, y);
}